// GaussianAttention_44100724195737
// MI455X (gfx1250) — compile-verified
//
#include <hip/hip_runtime.h>
#include <hip/hip_bf16.h>
#include <math.h>

#define HIDDEN 512
#define KMIX   10
#define BATCH  256
#define TLEN   2048
#define CDIM   128
#define TSPLIT 4

typedef __attribute__((ext_vector_type(2))) float v2f;
typedef __attribute__((ext_vector_type(4))) float v4f;
typedef __attribute__((ext_vector_type(8))) float v8f;

// ---------------------------------------------------------------------------
// Kernel 1: params = exp(h_t @ W + b); kappa += k_tm1.
// One wave (32 threads) computes one 16x16 output tile with
// V_WMMA_F32_16X16X4_F32 over K=512 (128 wmma steps).
// Column padding (30 -> 32) handled branch-free: clamped index + 0/1 mask,
// so the K-loop has no EXEC manipulation around the B loads.
// ---------------------------------------------------------------------------
__global__ void __launch_bounds__(32)
ga_params_wmma(const float* __restrict__ h_t,     // [B, HIDDEN]
               const float* __restrict__ k_tm1,   // [B, KMIX]
               const float* __restrict__ W,       // [HIDDEN, 3*KMIX]
               const float* __restrict__ bias,    // [3*KMIX]
               float* __restrict__ P_ws,          // [B, 32]
               float* __restrict__ out_kappa)     // [B, KMIX]
{
    const int lane  = threadIdx.x;         // 0..31 (wave32)
    const int m0    = blockIdx.x * 16;     // batch-row tile
    const int n0    = blockIdx.y * 16;     // output-col tile (0 or 16)

    const int mi    = lane & 15;           // A: M index / B: N index
    const int khalf = lane >> 4;           // 0 -> K{0,1}, 1 -> K{2,3}

    const int   gn   = n0 + mi;                            // this lane's B column
    const int   gnc  = (gn < 3 * KMIX) ? gn : (3 * KMIX - 1);  // clamped (valid addr)
    const float msk  = (gn < 3 * KMIX) ? 1.0f : 0.0f;      // zero out pad columns

    v8f acc = {};

    const float* arow = h_t + (size_t)(m0 + mi) * HIDDEN;
    const float* bcol = W + gnc;

    for (int kk = 0; kk < HIDDEN; kk += 4) {
        const int k0 = kk + khalf * 2;
        // A 16x4 tile: lane holds A[mi][k0], A[mi][k0+1] (contiguous -> b64)
        v2f a = *(const v2f*)(arow + k0);
        // B 4x16 tile: lane holds W[k0][gn], W[k0+1][gn]  (branch-free)
        v2f b;
        b.x = msk * bcol[(size_t)k0 * (3 * KMIX)];
        b.y = msk * bcol[(size_t)(k0 + 1) * (3 * KMIX)];
        acc = __builtin_amdgcn_wmma_f32_16x16x4_f32(
            /*neg_a=*/false, a, /*neg_b=*/false, b,
            /*c_mod=*/(short)0, acc, /*reuse_a=*/false, /*reuse_b=*/false);
    }

    // D layout: VGPR r holds M = r (lanes 0-15) / M = r+8 (lanes 16-31), N = lane&15.
    if (gn < 3 * KMIX) {
        const float bn = bias[gn];
        #pragma unroll
        for (int r = 0; r < 8; ++r) {
            const int gm = m0 + r + 8 * khalf;
            float v = __expf(acc[r] + bn);
            if (gn >= 2 * KMIX) {                 // kappa third
                v += k_tm1[gm * KMIX + (gn - 2 * KMIX)];
                out_kappa[gm * KMIX + (gn - 2 * KMIX)] = v;
            }
            P_ws[gm * 32 + gn] = v;
        }
    }
}

// ---------------------------------------------------------------------------
// Kernel 2: phi[b,t] = sum_k alpha_k * exp(-beta_k * (kappa_k - t)^2)
// grid (B, T/256), block 256. Params broadcast via LDS.
// ---------------------------------------------------------------------------
__global__ void __launch_bounds__(256)
ga_phi(const float* __restrict__ P_ws,   // [B, 32]
       float* __restrict__ out_phi)      // [B, T]
{
    __shared__ float sp[32];
    const int b = blockIdx.x;
    const int t = blockIdx.y * 256 + threadIdx.x;

    if (threadIdx.x < 32) sp[threadIdx.x] = P_ws[b * 32 + threadIdx.x];
    __syncthreads();

    const float ft = (float)t;
    float acc = 0.0f;
    #pragma unroll
    for (int j = 0; j < KMIX; ++j) {
        const float d = sp[2 * KMIX + j] - ft;     // kappa_j - t
        acc += sp[j] * __expf(-sp[KMIX + j] * d * d);
    }
    out_phi[b * TLEN + t] = acc;
}

// ---------------------------------------------------------------------------
// Kernel 3a: partial w over a T-chunk.
// grid (B, nchunk), block 256 = 8 t-slices x 32 lanes x float4 (C=128, b128
// non-temporal loads: ctx is a 268 MB single-use stream, keep it out of L2).
// With nchunk==1 this writes the final w directly (fallback path).
// ---------------------------------------------------------------------------
__global__ void __launch_bounds__(256)
ga_reduce_part(const float* __restrict__ ctx,     // [B, T, C]
               const float* __restrict__ phi,     // [B, T]
               float* __restrict__ part)          // [B, nchunk, C]
{
    const int b      = blockIdx.x;
    const int chunk  = blockIdx.y;
    const int nchunk = gridDim.y;
    const int rows   = TLEN / nchunk;
    const int t0     = chunk * rows;

    const int c4 = threadIdx.x & 31;   // float4 group: channels 4*c4..4*c4+3
    const int sl = threadIdx.x >> 5;   // t-slice 0..7

    const v4f* __restrict__ ctx4 = (const v4f*)(ctx + (size_t)b * TLEN * CDIM);
    const float* __restrict__ ph = phi + (size_t)b * TLEN;

    v4f acc = {};

    #pragma unroll 4
    for (int t = t0 + sl; t < t0 + rows; t += 8) {
        const float p = ph[t];
        const v4f v = __builtin_nontemporal_load(&ctx4[(size_t)t * (CDIM / 4) + c4]);
        acc += p * v;
    }

    __shared__ v4f red[8][32];
    red[sl][c4] = acc;
    __syncthreads();

    if (sl == 0) {
        v4f s = red[0][c4];
        #pragma unroll
        for (int i = 1; i < 8; ++i) s += red[i][c4];
        v4f* out4 = (v4f*)(part + ((size_t)b * nchunk + chunk) * CDIM);
        out4[c4] = s;
    }
}

// ---------------------------------------------------------------------------
// Kernel 3b: fold the nchunk partials into w. Tiny (512 KB traffic).
// ---------------------------------------------------------------------------
__global__ void __launch_bounds__(128)
ga_reduce_final(const float* __restrict__ part,   // [B, nchunk, C]
                float* __restrict__ out_w,        // [B, C]
                int nchunk)
{
    const int b = blockIdx.x;
    const int c = threadIdx.x;
    float s = 0.0f;
    for (int j = 0; j < nchunk; ++j)
        s += part[((size_t)b * nchunk + j) * CDIM + c];
    out_w[(size_t)b * CDIM + c] = s;
}

// ---------------------------------------------------------------------------
extern "C" void kernel_launch(void* const* d_in, const int* in_sizes, int n_in,
                              void* d_out, int out_size, void* d_ws, size_t ws_size,
                              hipStream_t stream) {
    const float* h_t   = (const float*)d_in[0];   // [B, HIDDEN]
    const float* k_tm1 = (const float*)d_in[1];   // [B, 1, KMIX]
    const float* ctx   = (const float*)d_in[2];   // [B, T, C]
    const float* W     = (const float*)d_in[3];   // [HIDDEN, 3*KMIX]
    const float* bias  = (const float*)d_in[4];   // [3*KMIX]

    float* out_w     = (float*)d_out;                                  // [B, C]
    float* out_phi   = out_w + (size_t)BATCH * CDIM;                   // [B, T, 1]
    float* out_kappa = out_phi + (size_t)BATCH * TLEN;                 // [B, 1, KMIX]

    float* P_ws  = (float*)d_ws;                                       // [B, 32]
    float* part  = P_ws + (size_t)BATCH * 32;                          // [B, TSPLIT, C]

    const size_t ws_needed =
        (size_t)BATCH * 32 * sizeof(float) +
        (size_t)BATCH * TSPLIT * CDIM * sizeof(float);

    // K1: params GEMM via WMMA (16 m-tiles x 2 n-tiles, one wave each)
    ga_params_wmma<<<dim3(BATCH / 16, 2), 32, 0, stream>>>(
        h_t, k_tm1, W, bias, P_ws, out_kappa);

    // K2: phi
    ga_phi<<<dim3(BATCH, TLEN / 256), 256, 0, stream>>>(P_ws, out_phi);

    // K3: w = sum_t phi * ctx  (streams ctx once; split T for occupancy)
    if (ws_size >= ws_needed) {
        ga_reduce_part<<<dim3(BATCH, TSPLIT), 256, 0, stream>>>(ctx, out_phi, part);
        ga_reduce_final<<<dim3(BATCH), 128, 0, stream>>>(part, out_w, TSPLIT);
    } else {
        // Scratch too small for partials: single-chunk path writes w directly.
        ga_reduce_part<<<dim3(BATCH, 1), 256, 0, stream>>>(ctx, out_phi, out_w);
    }
}